// PatchMasker_17051020165372
// MI455X (gfx1250) — compile-verified
//
#include <hip/hip_runtime.h>
#include <hip/hip_bf16.h>

// Problem constants (from reference):
//   B=64, T=512, F=256, MASKING_RATE=0.4 -> n_mask = ceil(512*0.4) = 205
//   fill = 0.0f
#define T_DIM   512
#define F_DIM   256
#define N_ELEMS (64 * 512 * 256)          // per-tensor element count = 8,388,608
#define N_VEC4  (N_ELEMS / 4)             // float4 count = 2,097,152
#define N_MASK  205

// Native clang vector type: accepted by __builtin_nontemporal_* (HIP's float4
// struct is not), same 16-byte size/alignment -> global_load/store_b128.
typedef float v4f __attribute__((ext_vector_type(4)));

// ---------------------------------------------------------------------------
// Kernel 1: rank-selection mask.
// mask[t] = true iff random_tensor[t] is among the N_MASK smallest values,
// with stable-argsort tie-breaking (earlier index wins on equal values).
// One block of 512 threads; the 512-float vector lives in LDS.
// ---------------------------------------------------------------------------
__global__ __launch_bounds__(T_DIM)
void patchmask_rank_kernel(const float* __restrict__ r,
                           unsigned*    __restrict__ mask_ws,   // 512 x u32 (0/1)
                           float*       __restrict__ mask_out)  // 512 x f32 (0.0/1.0)
{
    __shared__ float s[T_DIM];
    const int t = threadIdx.x;
    s[t] = r[t];
    __syncthreads();

    const float v = s[t];
    int cnt = 0;
#pragma unroll 8
    for (int j = 0; j < T_DIM; ++j) {
        const float u = s[j];
        cnt += (u < v) || ((u == v) && (j < t));   // stable rank
    }
    const bool m = (cnt < N_MASK);
    mask_ws[t]  = m ? 1u   : 0u;
    mask_out[t] = m ? 1.0f : 0.0f;
}

// ---------------------------------------------------------------------------
// Kernel 2: streaming apply. Each thread owns one float4 of each of the three
// tensors. For flat float4 index i: row = i / 64 (F/4 = 64 float4 per row),
// t = row % 512. The mask is uniform across each wave32 (64 consecutive
// threads share t), so the branch is divergence-free and masked rows skip
// their global reads entirely (~40% read-traffic savings).
// Nontemporal (TH=NT) b128 loads/stores: streams >> 192MB L2, zero reuse.
// ---------------------------------------------------------------------------
__global__ __launch_bounds__(256)
void patchmask_apply_kernel(const v4f* __restrict__ xt,
                            const v4f* __restrict__ xs,
                            const v4f* __restrict__ xr,
                            const unsigned* __restrict__ mask,
                            v4f* __restrict__ zt,
                            v4f* __restrict__ zs,
                            v4f* __restrict__ zr)
{
    const int i = blockIdx.x * blockDim.x + threadIdx.x;   // 0 .. N_VEC4-1
    const int t = (i >> 6) & (T_DIM - 1);                  // time index

    const unsigned m = mask[t];                            // tiny, cache-resident

    v4f a, b, c;
    if (m != 0u) {
        a = (v4f)(0.0f);
        b = (v4f)(0.0f);
        c = (v4f)(0.0f);
    } else {
        a = __builtin_nontemporal_load(&xt[i]);
        b = __builtin_nontemporal_load(&xs[i]);
        c = __builtin_nontemporal_load(&xr[i]);
    }
    __builtin_nontemporal_store(a, &zt[i]);
    __builtin_nontemporal_store(b, &zs[i]);
    __builtin_nontemporal_store(c, &zr[i]);
}

extern "C" void kernel_launch(void* const* d_in, const int* in_sizes, int n_in,
                              void* d_out, int out_size, void* d_ws, size_t ws_size,
                              hipStream_t stream) {
    (void)in_sizes; (void)n_in; (void)out_size; (void)ws_size;

    const float* x_tre = (const float*)d_in[0];
    const float* x_sea = (const float*)d_in[1];
    const float* x_res = (const float*)d_in[2];
    const float* rnd   = (const float*)d_in[3];

    float* out   = (float*)d_out;
    float* z_tre = out;
    float* z_sea = out + (size_t)N_ELEMS;
    float* z_res = out + 2 * (size_t)N_ELEMS;
    float* m_out = out + 3 * (size_t)N_ELEMS;   // 512 floats (bool mask as 0/1)

    unsigned* mask_ws = (unsigned*)d_ws;        // 2 KB scratch

    // 1) compute the mask (one tiny block)
    patchmask_rank_kernel<<<1, T_DIM, 0, stream>>>(rnd, mask_ws, m_out);

    // 2) apply it to all three tensors (bandwidth-bound streaming)
    const int threads = 256;
    const int blocks  = N_VEC4 / threads;       // 8192
    patchmask_apply_kernel<<<blocks, threads, 0, stream>>>(
        (const v4f*)x_tre, (const v4f*)x_sea, (const v4f*)x_res,
        mask_ws,
        (v4f*)z_tre, (v4f*)z_sea, (v4f*)z_res);
}